// GIN_53137335386821
// MI455X (gfx1250) — compile-verified
//
#include <hip/hip_runtime.h>

typedef float v2f __attribute__((ext_vector_type(2)));
typedef float v8f __attribute__((ext_vector_type(8)));

static inline size_t cdiv_sz(size_t a, size_t b) { return (a + b - 1) / b; }

// ---------------------------------------------------------------------------
// gfx1250 async global->LDS copy helpers (ASYNCcnt-tracked, ISA ch.10/15.18)
// ---------------------------------------------------------------------------
__device__ __forceinline__ unsigned lds_off_of(const void* p) {
  // addrspacecast(3->flat) puts the LDS byte offset in addr[31:0]
  return (unsigned)(unsigned long long)p;
}

__device__ __forceinline__ void async_load_b128(unsigned lds_off, const float* gaddr) {
  asm volatile("global_load_async_to_lds_b128 %0, %1, off"
               :: "v"(lds_off), "v"(gaddr)
               : "memory");
}

__device__ __forceinline__ void s_wait_async3() {
  asm volatile("s_wait_asynccnt 0x3" ::: "memory");
}
__device__ __forceinline__ void s_wait_async0() {
  asm volatile("s_wait_asynccnt 0x0" ::: "memory");
}

// ---------------------------------------------------------------------------
// Elementwise copy (float4 vectorized)
// ---------------------------------------------------------------------------
__global__ __launch_bounds__(256) void copy_f4(const float* __restrict__ in,
                                               float* __restrict__ out, size_t n4) {
  size_t i = (size_t)blockIdx.x * blockDim.x + threadIdx.x;
  if (i < n4) ((float4*)out)[i] = ((const float4*)in)[i];
}

// ---------------------------------------------------------------------------
// GIN aggregation: out[dst] += h[src] over all edges (out pre-initialized = h)
// ---------------------------------------------------------------------------
__global__ __launch_bounds__(256) void edge_agg(const int* __restrict__ ei,
                                                const float* __restrict__ h,
                                                float* __restrict__ out,
                                                int E, int F) {
  const int F4 = F >> 2;
  size_t i = (size_t)blockIdx.x * blockDim.x + threadIdx.x;
  size_t total = (size_t)E * F4;
  if (i >= total) return;
  int e = (int)(i / F4);
  int f = (int)(i % F4) << 2;
  int src = ei[e];
  int dst = ei[E + e];
  float4 v = *(const float4*)(h + (size_t)src * F + f);
  float* o = out + (size_t)dst * F + f;
  atomicAdd(o + 0, v.x);
  atomicAdd(o + 1, v.y);
  atomicAdd(o + 2, v.z);
  atomicAdd(o + 3, v.w);
}

// ---------------------------------------------------------------------------
// f32 WMMA GEMM: C[M,Nc] = op(A[M,K] @ W[K,Nc] + bias)
// Block: 256 threads = 8 waves; tile 64(M) x 128(N); K in 16-chunks,
// double-buffered via GLOBAL_LOAD_ASYNC_TO_LDS_B128 (ASYNCcnt).
// Each wave: 4 accumulators (16x16 each), A-frag reused across 4 N-tiles.
// Requires: K % 16 == 0, Nc % 128 == 0.
// ---------------------------------------------------------------------------
template <bool RELU>
__global__ __launch_bounds__(256) void gemm_f32_wmma(const float* __restrict__ A,
                                                     const float* __restrict__ W,
                                                     const float* __restrict__ bias,
                                                     float* __restrict__ C,
                                                     int M, int K, int Nc) {
  __shared__ float As[2][64][20];   // 64 rows x 16 K  (+4 pad; 80B pitch, 16B aligned)
  __shared__ float Bs[2][16][132];  // 16 K x 128 cols (+4 pad; 528B pitch, 16B aligned)

  const int tid = threadIdx.x;
  const int lane = tid & 31;
  const int wave = tid >> 5;    // 0..7
  const int waveM = wave & 3;   // 16-row subtile
  const int waveN = wave >> 2;  // 64-col group (0..1)
  const int rowBlock = blockIdx.x * 64;
  const int colBlock = blockIdx.y * 128;

  const int m = lane & 15;
  const int koff = (lane >> 4) * 2;  // lanes 16-31 carry K+2/K+3 (ISA 7.12.2)

  // A staging: 64x16 floats, one b128 per thread. Clamp tail rows (keeps EXEC
  // full + uniform ASYNCcnt; duplicated rows only feed never-stored outputs).
  const int aRow = tid >> 2;
  const int aCol = (tid & 3) * 4;
  int aGRow = rowBlock + aRow;
  if (aGRow >= M) aGRow = M - 1;
  const float* aBase = A + (size_t)aGRow * K + aCol;

  // B staging: 16x128 floats, two b128 per thread.
  const int bRow0 = tid >> 5;          // 0..7
  const int bRow1 = (tid + 256) >> 5;  // 8..15
  const int bCol = (tid & 31) * 4;     // 0..124
  const float* bBase0 = W + (size_t)bRow0 * Nc + colBlock + bCol;
  const float* bBase1 = W + (size_t)bRow1 * Nc + colBlock + bCol;

  auto issue_chunk = [&](int kc, int buf) {
    async_load_b128(lds_off_of(&As[buf][aRow][aCol]), aBase + kc);
    async_load_b128(lds_off_of(&Bs[buf][bRow0][bCol]), bBase0 + (size_t)kc * Nc);
    async_load_b128(lds_off_of(&Bs[buf][bRow1][bCol]), bBase1 + (size_t)kc * Nc);
  };

  const v8f vzero = {0.f, 0.f, 0.f, 0.f, 0.f, 0.f, 0.f, 0.f};
  v8f acc[4] = {vzero, vzero, vzero, vzero};

  const int nch = K >> 4;
  issue_chunk(0, 0);
  int cur = 0;

  for (int i = 0; i < nch; ++i) {
    if (i + 1 < nch) {
      __syncthreads();                   // buf[cur^1] free (prev compute done)
      issue_chunk((i + 1) << 4, cur ^ 1);
      s_wait_async3();                   // chunk i's 3 async ops retired (in-order)
    } else {
      s_wait_async0();
    }
    __syncthreads();                     // publish all waves' LDS writes

#pragma unroll
    for (int k4 = 0; k4 < 16; k4 += 4) {
      v2f a;
      a.x = As[cur][waveM * 16 + m][k4 + koff];
      a.y = As[cur][waveM * 16 + m][k4 + koff + 1];
#pragma unroll
      for (int t = 0; t < 4; ++t) {
        v2f b;
        b.x = Bs[cur][k4 + koff][waveN * 64 + t * 16 + m];
        b.y = Bs[cur][k4 + koff + 1][waveN * 64 + t * 16 + m];
        acc[t] = __builtin_amdgcn_wmma_f32_16x16x4_f32(false, a, false, b,
                                                       (short)0, acc[t], false, false);
      }
    }
    cur ^= 1;
  }

  // Epilogue: C/D layout — VGPR r: lanes0-15 -> M=r, lanes16-31 -> M=r+8
  const int rbase = rowBlock + waveM * 16 + ((lane >> 4) * 8);
#pragma unroll
  for (int t = 0; t < 4; ++t) {
    const int col = colBlock + waveN * 64 + t * 16 + m;
    const float bv = bias[col];
#pragma unroll
    for (int r = 0; r < 8; ++r) {
      int row = rbase + r;
      if (row < M) {
        float v = acc[t][r] + bv;
        if (RELU) v = fmaxf(v, 0.f);
        C[(size_t)row * Nc + col] = v;
      }
    }
  }
}

// ---------------------------------------------------------------------------
// BatchNorm (training mode) over node dim: column sums / sums of squares
// ---------------------------------------------------------------------------
__global__ __launch_bounds__(256) void col_stats(const float* __restrict__ h,
                                                 float* __restrict__ sum,
                                                 float* __restrict__ sumsq,
                                                 int N, int H) {
  int tid = blockIdx.x * blockDim.x + threadIdx.x;  // H * 128 threads
  int c = tid % H;
  int chunk = tid / H;  // 0..127
  float s = 0.f, s2 = 0.f;
  for (int r = chunk; r < N; r += 128) {
    float v = h[(size_t)r * H + c];
    s += v;
    s2 += v * v;
  }
  atomicAdd(&sum[c], s);
  atomicAdd(&sumsq[c], s2);
}

__global__ __launch_bounds__(256) void bn_finalize(const float* __restrict__ sum,
                                                   const float* __restrict__ sumsq,
                                                   const float* __restrict__ g,
                                                   const float* __restrict__ bt,
                                                   float* __restrict__ scale,
                                                   float* __restrict__ shift,
                                                   int H, float invN) {
  int c = blockIdx.x * blockDim.x + threadIdx.x;
  if (c >= H) return;
  float mean = sum[c] * invN;
  float var = sumsq[c] * invN - mean * mean;  // biased var (jnp.var default)
  float inv = rsqrtf(var + 1e-5f);
  float sc = g[c] * inv;
  scale[c] = sc;
  shift[c] = bt[c] - mean * sc;
}

// ---------------------------------------------------------------------------
// Fused: h = relu(h*scale+shift) in place  +  global_add_pool into P
// ---------------------------------------------------------------------------
__global__ __launch_bounds__(256) void bn_relu_pool(float* __restrict__ h,
                                                    const float* __restrict__ scale,
                                                    const float* __restrict__ shift,
                                                    const int* __restrict__ batch,
                                                    float* __restrict__ P,
                                                    int N, int H, int off, int R) {
  const int H4 = H >> 2;
  size_t i = (size_t)blockIdx.x * blockDim.x + threadIdx.x;
  size_t total = (size_t)N * H4;
  if (i >= total) return;
  int n = (int)(i / H4);
  int f = (int)(i % H4) << 2;
  float4 v = *(const float4*)(h + (size_t)n * H + f);
  const float4 sc = *(const float4*)(scale + f);
  const float4 sh = *(const float4*)(shift + f);
  v.x = fmaxf(v.x * sc.x + sh.x, 0.f);
  v.y = fmaxf(v.y * sc.y + sh.y, 0.f);
  v.z = fmaxf(v.z * sc.z + sh.z, 0.f);
  v.w = fmaxf(v.w * sc.w + sh.w, 0.f);
  *(float4*)(h + (size_t)n * H + f) = v;
  int g = batch[n];
  float* o = P + (size_t)g * R + off + f;
  atomicAdd(o + 0, v.x);
  atomicAdd(o + 1, v.y);
  atomicAdd(o + 2, v.z);
  atomicAdd(o + 3, v.w);
}

// ---------------------------------------------------------------------------
// Host orchestration
// ---------------------------------------------------------------------------
extern "C" void kernel_launch(void* const* d_in, const int* in_sizes, int n_in,
                              void* d_out, int out_size, void* d_ws, size_t ws_size,
                              hipStream_t stream) {
  const float* x   = (const float*)d_in[0];
  const int*   ei  = (const int*)d_in[1];
  const int*   bat = (const int*)d_in[2];
  const float* w1a = (const float*)d_in[3];  const float* b1a = (const float*)d_in[4];
  const float* w1b = (const float*)d_in[5];  const float* b1b = (const float*)d_in[6];
  const float* g1  = (const float*)d_in[7];  const float* bt1 = (const float*)d_in[8];
  const float* w2a = (const float*)d_in[9];  const float* b2a = (const float*)d_in[10];
  const float* w2b = (const float*)d_in[11]; const float* b2b = (const float*)d_in[12];
  const float* g2  = (const float*)d_in[13]; const float* bt2 = (const float*)d_in[14];
  const float* w3  = (const float*)d_in[15]; const float* b3  = (const float*)d_in[16];
  const float* g3  = (const float*)d_in[17]; const float* bt3 = (const float*)d_in[18];
  const float* wl1 = (const float*)d_in[19]; const float* bl1 = (const float*)d_in[20];
  const float* wl2 = (const float*)d_in[21]; const float* bl2 = (const float*)d_in[22];

  const int N  = in_sizes[2];
  const int E  = in_sizes[1] / 2;
  const int F  = in_sizes[0] / N;   // 64
  const int H  = in_sizes[4];       // 128
  const int H3 = in_sizes[16];      // 512
  const int L1 = in_sizes[20];      // 1024
  const int C  = in_sizes[22];      // 256
  const int R  = H + H + H3;        // 768
  const int G  = 256;               // NUM_GRAPHS

  // Workspace carve-out
  const size_t Npad = ((size_t)N + 63) & ~(size_t)63;
  char* ws = (char*)d_ws;
  size_t off = 0;
  auto carve = [&](size_t bytes) -> float* {
    float* p = (float*)(ws + off);
    off = (off + bytes + 255) & ~(size_t)255;
    return p;
  };
  float* bufA  = carve(Npad * H * 4);   // aggregation output (F or H cols)
  float* bufT  = carve(Npad * H * 4);   // MLP hidden
  float* h1    = carve(Npad * H * 4);
  float* h2    = carve(Npad * H * 4);
  float* h3    = carve(Npad * H3 * 4);
  float* P     = carve((size_t)G * R * 4);
  float* r1    = carve((size_t)G * L1 * 4);
  float* sum   = carve(512 * 4);
  float* sumsq = carve(512 * 4);
  float* scale = carve(512 * 4);
  float* shift = carve(512 * 4);
  (void)ws_size; (void)n_in; (void)out_size;

  const dim3 blk(256);

  (void)hipMemsetAsync(P, 0, (size_t)G * R * 4, stream);

  auto run_gemm = [&](const float* A, const float* W, const float* bias, float* Cout,
                      int M, int K, int Nc, bool relu) {
    dim3 grid((unsigned)cdiv_sz(M, 64), (unsigned)(Nc / 128));
    if (relu)
      gemm_f32_wmma<true><<<grid, blk, 0, stream>>>(A, W, bias, Cout, M, K, Nc);
    else
      gemm_f32_wmma<false><<<grid, blk, 0, stream>>>(A, W, bias, Cout, M, K, Nc);
  };

  auto run_agg = [&](const float* hin, float* aout, int Fc) {
    size_t n4 = (size_t)N * Fc / 4;
    copy_f4<<<dim3((unsigned)cdiv_sz(n4, 256)), blk, 0, stream>>>(hin, aout, n4);
    size_t tot = (size_t)E * (Fc / 4);
    edge_agg<<<dim3((unsigned)cdiv_sz(tot, 256)), blk, 0, stream>>>(ei, hin, aout, E, Fc);
  };

  auto run_bn_pool = [&](float* hbuf, const float* g, const float* bt, int Hc, int offc) {
    (void)hipMemsetAsync(sum, 0, Hc * 4, stream);
    (void)hipMemsetAsync(sumsq, 0, Hc * 4, stream);
    col_stats<<<dim3((unsigned)(Hc * 128 / 256)), blk, 0, stream>>>(hbuf, sum, sumsq, N, Hc);
    bn_finalize<<<dim3((unsigned)cdiv_sz(Hc, 256)), blk, 0, stream>>>(
        sum, sumsq, g, bt, scale, shift, Hc, 1.0f / (float)N);
    size_t tot = (size_t)N * (Hc / 4);
    bn_relu_pool<<<dim3((unsigned)cdiv_sz(tot, 256)), blk, 0, stream>>>(
        hbuf, scale, shift, bat, P, N, Hc, offc, R);
  };

  // ---- Layer 1: h1 = relu(BN(relu(agg(x)@w1a+b1a)@w1b+b1b)) ----
  run_agg(x, bufA, F);
  run_gemm(bufA, w1a, b1a, bufT, N, F, H, true);
  run_gemm(bufT, w1b, b1b, h1, N, H, H, false);
  run_bn_pool(h1, g1, bt1, H, 0);

  // ---- Layer 2 ----
  run_agg(h1, bufA, H);
  run_gemm(bufA, w2a, b2a, bufT, N, H, H, true);
  run_gemm(bufT, w2b, b2b, h2, N, H, H, false);
  run_bn_pool(h2, g2, bt2, H, H);

  // ---- Layer 3: h3 = relu(BN(relu(agg(h2)@w3+b3))) ----
  run_agg(h2, bufA, H);
  run_gemm(bufA, w3, b3, h3, N, H, H3, true);
  run_bn_pool(h3, g3, bt3, H3, 2 * H);

  // ---- Readout MLP: out = relu(P@wl1+bl1)@wl2+bl2 ----
  run_gemm(P, wl1, bl1, r1, G, R, L1, true);
  run_gemm(r1, wl2, bl2, (float*)d_out, G, L1, C, false);
}